// VIGraph_19808389169187
// MI455X (gfx1250) — compile-verified
//
#include <hip/hip_runtime.h>
#include <hip/hip_bf16.h>
#include <math.h>

// ---------------------------------------------------------------------------
// VI pose-graph residuals (float32). Memory-bound; no matrix structure.
//   out = concat( L1 * se3_log(vmot^-1 * (pose1^-1 * pose2))   [6E]
//               , L2 * (qrot(r1,body_dvel) - diff(vel))        [3(N-1)]
//               , L3 * so3_log(imu_drot * conj(r1) * r2)       [3(N-1)]
//               , L4 * (qrot(r1,body_dpos)+vel*DT - diff(t))   [3(N-1)] )
// ---------------------------------------------------------------------------

#define DT   0.1f
#define L1c  1.0f
#define L2c  0.5f
#define L3c  2.0f
#define L4c  0.5f
#define EPSc 1e-06f

typedef float v4f __attribute__((ext_vector_type(4)));

struct F3 { float x, y, z; };
struct Q4 { float x, y, z, w; };   // v = (x,y,z), scalar w

__device__ __forceinline__ F3 f3(float a, float b, float c) { F3 r{a,b,c}; return r; }
__device__ __forceinline__ F3 add3(F3 a, F3 b) { return f3(a.x+b.x, a.y+b.y, a.z+b.z); }
__device__ __forceinline__ F3 sub3(F3 a, F3 b) { return f3(a.x-b.x, a.y-b.y, a.z-b.z); }
__device__ __forceinline__ F3 scl3(float s, F3 a) { return f3(s*a.x, s*a.y, s*a.z); }
__device__ __forceinline__ float dot3(F3 a, F3 b) { return fmaf(a.x,b.x, fmaf(a.y,b.y, a.z*b.z)); }
__device__ __forceinline__ F3 cross3(F3 a, F3 b) {
    return f3(a.y*b.z - a.z*b.y,
              a.z*b.x - a.x*b.z,
              a.x*b.y - a.y*b.x);
}

// qrot: x + 2*cross(v, cross(v,x) + w*x)
__device__ __forceinline__ F3 qrot(Q4 q, F3 p) {
    F3 v = f3(q.x, q.y, q.z);
    F3 t = add3(cross3(v, p), scl3(q.w, p));
    return add3(p, scl3(2.0f, cross3(v, t)));
}

__device__ __forceinline__ Q4 qmul(Q4 a, Q4 b) {
    F3 v1 = f3(a.x,a.y,a.z), v2 = f3(b.x,b.y,b.z);
    F3 v = add3(add3(scl3(a.w, v2), scl3(b.w, v1)), cross3(v1, v2));
    Q4 r; r.x = v.x; r.y = v.y; r.z = v.z;
    r.w = a.w*b.w - dot3(v1, v2);
    return r;
}

__device__ __forceinline__ Q4 qconj(Q4 q) { Q4 r{-q.x,-q.y,-q.z,q.w}; return r; }

// so3_log with the reference's EPS selects
__device__ __forceinline__ F3 so3_log(Q4 q) {
    F3 v = f3(q.x, q.y, q.z);
    float n = sqrtf(dot3(v, v));
    float theta  = 2.0f * atan2f(n, q.w);
    float w_safe = (fabsf(q.w) < EPSc) ? 1.0f : q.w;
    float n_safe = (n < EPSc) ? 1.0f : n;
    float factor = (n < EPSc) ? (2.0f / w_safe) : (theta / n_safe);
    return scl3(factor, v);
}

// ---------------------------------------------------------------------------
// Edge kernel: one lane per edge. pose gathers hit L2 (pose = 7.3MB, RT hint);
// vmot / vlink / output are pure streams -> non-temporal so they do not evict
// the pose working set from the 192MB L2.
// ---------------------------------------------------------------------------
__global__ __launch_bounds__(256) void vigraph_edges(
    const float* __restrict__ pose,   // N x 7
    const float* __restrict__ vmot,   // E x 7
    const int*   __restrict__ vlink,  // E x 2
    float*       __restrict__ out,    // 6E  (L1 * pgerr)
    int E)
{
    int e = blockIdx.x * blockDim.x + threadIdx.x;
    if (e >= E) return;

    int i1 = __builtin_nontemporal_load(vlink + 2*e + 0);
    int i2 = __builtin_nontemporal_load(vlink + 2*e + 1);

    const float* p1p = pose + 7LL*i1;
    const float* p2p = pose + 7LL*i2;
    F3 t1 = f3(p1p[0], p1p[1], p1p[2]);
    Q4 q1; q1.x=p1p[3]; q1.y=p1p[4]; q1.z=p1p[5]; q1.w=p1p[6];
    F3 t2 = f3(p2p[0], p2p[1], p2p[2]);
    Q4 q2; q2.x=p2p[3]; q2.y=p2p[4]; q2.z=p2p[5]; q2.w=p2p[6];

    const float* vmp = vmot + 7LL*e;
    F3 tv = f3(__builtin_nontemporal_load(vmp+0),
               __builtin_nontemporal_load(vmp+1),
               __builtin_nontemporal_load(vmp+2));
    Q4 qv; qv.x=__builtin_nontemporal_load(vmp+3);
           qv.y=__builtin_nontemporal_load(vmp+4);
           qv.z=__builtin_nontemporal_load(vmp+5);
           qv.w=__builtin_nontemporal_load(vmp+6);

    // A = se3_inv(pose1) * pose2
    Q4 qi1 = qconj(q1);
    F3 ti1 = scl3(-1.0f, qrot(qi1, t1));
    F3 tA  = add3(ti1, qrot(qi1, t2));
    Q4 qA  = qmul(qi1, q2);

    // err = se3_inv(vmot) * A
    Q4 qiv = qconj(qv);
    F3 tiv = scl3(-1.0f, qrot(qiv, tv));
    F3 te  = add3(tiv, qrot(qiv, tA));
    Q4 qe  = qmul(qiv, qA);

    // se3_log(err)
    F3 phi = so3_log(qe);
    float th = sqrtf(dot3(phi, phi));
    float th_safe = (th < EPSc) ? 1.0f : th;
    float coef;
    if (th < EPSc) {
        coef = (1.0f/12.0f) + th*th*(1.0f/720.0f);
    } else {
        coef = 1.0f/(th_safe*th_safe)
             - (1.0f + cosf(th_safe)) / (2.0f * th_safe * sinf(th_safe));
    }
    F3 pxt = cross3(phi, te);
    F3 tau = add3(sub3(te, scl3(0.5f, pxt)), scl3(coef, cross3(phi, pxt)));

    float* o = out + 6LL*e;
    __builtin_nontemporal_store(L1c * tau.x, o+0);
    __builtin_nontemporal_store(L1c * tau.y, o+1);
    __builtin_nontemporal_store(L1c * tau.z, o+2);
    __builtin_nontemporal_store(L1c * phi.x, o+3);
    __builtin_nontemporal_store(L1c * phi.y, o+4);
    __builtin_nontemporal_store(L1c * phi.z, o+5);
}

// ---------------------------------------------------------------------------
// Chain kernel: one lane per timestep i in [0, N-1). Sequential streams.
// ---------------------------------------------------------------------------
__global__ __launch_bounds__(256) void vigraph_chain(
    const float* __restrict__ pose,       // N x 7
    const float* __restrict__ vel,        // N x 3
    const float* __restrict__ imu_drot,   // (N-1) x 4 (16B-aligned rows)
    const float* __restrict__ body_dvel,  // (N-1) x 3
    const float* __restrict__ body_dpos,  // (N-1) x 3
    float*       __restrict__ out,        // 9(N-1): [adjvel | imurot | transvel]
    int Nm1)
{
    int i = blockIdx.x * blockDim.x + threadIdx.x;
    if (i >= Nm1) return;

    const float* pA = pose + 7LL*i;
    const float* pB = pose + 7LL*(i+1);
    F3 tr1 = f3(pA[0], pA[1], pA[2]);
    Q4 r1;  r1.x=pA[3]; r1.y=pA[4]; r1.z=pA[5]; r1.w=pA[6];
    F3 tr2 = f3(pB[0], pB[1], pB[2]);
    Q4 r2;  r2.x=pB[3]; r2.y=pB[4]; r2.z=pB[5]; r2.w=pB[6];

    F3 v1 = f3(vel[3LL*i+0],     vel[3LL*i+1],     vel[3LL*i+2]);
    F3 v2 = f3(vel[3LL*(i+1)+0], vel[3LL*(i+1)+1], vel[3LL*(i+1)+2]);

    F3 bdv = f3(__builtin_nontemporal_load(body_dvel + 3LL*i+0),
                __builtin_nontemporal_load(body_dvel + 3LL*i+1),
                __builtin_nontemporal_load(body_dvel + 3LL*i+2));
    F3 bdp = f3(__builtin_nontemporal_load(body_dpos + 3LL*i+0),
                __builtin_nontemporal_load(body_dpos + 3LL*i+1),
                __builtin_nontemporal_load(body_dpos + 3LL*i+2));

    // imu_drot rows are 16B-aligned: single NT b128 load via native clang vector
    v4f dr4 = __builtin_nontemporal_load((const v4f*)(imu_drot) + i);
    Q4 drot; drot.x=dr4.x; drot.y=dr4.y; drot.z=dr4.z; drot.w=dr4.w;

    // adjvelerr = qrot(r1, body_dvel) - (vel[i+1]-vel[i])
    F3 adv = sub3(qrot(r1, bdv), sub3(v2, v1));

    // imuroterr = so3_log( imu_drot * (conj(r1) * r2) )
    Q4 qrel = qmul(drot, qmul(qconj(r1), r2));
    F3 ire  = so3_log(qrel);

    // transvelerr = qrot(r1, body_dpos) + vel[i]*DT - (trans[i+1]-trans[i])
    F3 tve = sub3(add3(qrot(r1, bdp), scl3(DT, v1)), sub3(tr2, tr1));

    float* o0 = out + 3LL*i;                   // L2 * adjvelerr
    float* o1 = out + 3LL*(Nm1 + i);           // L3 * imuroterr
    float* o2 = out + 3LL*(2LL*Nm1 + i);       // L4 * transvelerr
    __builtin_nontemporal_store(L2c * adv.x, o0+0);
    __builtin_nontemporal_store(L2c * adv.y, o0+1);
    __builtin_nontemporal_store(L2c * adv.z, o0+2);
    __builtin_nontemporal_store(L3c * ire.x, o1+0);
    __builtin_nontemporal_store(L3c * ire.y, o1+1);
    __builtin_nontemporal_store(L3c * ire.z, o1+2);
    __builtin_nontemporal_store(L4c * tve.x, o2+0);
    __builtin_nontemporal_store(L4c * tve.y, o2+1);
    __builtin_nontemporal_store(L4c * tve.z, o2+2);
}

extern "C" void kernel_launch(void* const* d_in, const int* in_sizes, int n_in,
                              void* d_out, int out_size, void* d_ws, size_t ws_size,
                              hipStream_t stream) {
    const float* pose      = (const float*)d_in[0]; // N*7
    const float* vel       = (const float*)d_in[1]; // N*3
    const float* vmot      = (const float*)d_in[2]; // E*7
    const float* imu_drot  = (const float*)d_in[3]; // (N-1)*4
    const float* body_dvel = (const float*)d_in[4]; // (N-1)*3
    const float* body_dpos = (const float*)d_in[5]; // (N-1)*3
    const int*   vlink     = (const int*)  d_in[6]; // E*2

    const int N   = in_sizes[0] / 7;
    const int E   = in_sizes[2] / 7;
    const int Nm1 = N - 1;

    float* out = (float*)d_out;

    const int TPB = 256;
    vigraph_edges<<<(E + TPB - 1) / TPB, TPB, 0, stream>>>(
        pose, vmot, vlink, out, E);

    vigraph_chain<<<(Nm1 + TPB - 1) / TPB, TPB, 0, stream>>>(
        pose, vel, imu_drot, body_dvel, body_dpos, out + 6LL*E, Nm1);
}